// GNNDecoder_26242250179179
// MI455X (gfx1250) — compile-verified
//
#include <hip/hip_runtime.h>
#include <math.h>

typedef float v2f __attribute__((ext_vector_type(2)));
typedef float v8f __attribute__((ext_vector_type(8)));

#define N_NODES 1024
#define D 128

// ---------------------------------------------------------------------------
// m_out[j] = b[j] + sum_k ( (1/N) * sum_r X[r][k] ) * W[k][j]
// One block, 1024 threads (32 waves). Column sums via 8-way LDS reduction.
// ---------------------------------------------------------------------------
__global__ __launch_bounds__(1024) void mean_matvec_kernel(
    const float* __restrict__ X, const float* __restrict__ W,
    const float* __restrict__ b, float* __restrict__ m_out)
{
    __shared__ float part[8 * D];
    __shared__ float meanv[D];
    const int tid = threadIdx.x;
    const int c = tid & (D - 1);   // column 0..127
    const int g = tid >> 7;        // row group 0..7

    float acc = 0.f;
    for (int r = g; r < N_NODES; r += 8)
        acc += X[r * D + c];
    part[g * D + c] = acc;
    __syncthreads();

    if (g == 0) {
        float s = 0.f;
#pragma unroll
        for (int gg = 0; gg < 8; ++gg) s += part[gg * D + c];
        meanv[c] = s * (1.0f / (float)N_NODES);
    }
    __syncthreads();

    if (tid < D) {
        float a = b[tid];
        for (int k = 0; k < D; ++k)
            a += meanv[k] * W[k * D + tid];
        m_out[tid] = a;
    }
}

// ---------------------------------------------------------------------------
// out[r][c] = X[r][c] + m[c]   (optionally ReLU)
// ---------------------------------------------------------------------------
__global__ __launch_bounds__(256) void add_bias_kernel(
    const float* __restrict__ X, const float* __restrict__ m,
    float* __restrict__ out, int do_relu)
{
    const int idx = blockIdx.x * 256 + threadIdx.x;   // < N_NODES*D
    const int c = idx & (D - 1);
    float v = X[idx] + m[c];
    if (do_relu) v = fmaxf(v, 0.f);
    out[idx] = v;
}

// ---------------------------------------------------------------------------
// G = H * H^T using V_WMMA_F32_16X16X4_F32. One wave per 16x16 output tile.
// A 16x4 layout: lanes 0-15 -> K = k0,k0+1 ; lanes 16-31 -> K = k0+2,k0+3.
// B 4x16 layout mirrors it with N = lane&15. For the Gram matrix both A and B
// per-lane data is a float2 from H[row][k0 + 2*(lane>=16)].
// ---------------------------------------------------------------------------
__global__ __launch_bounds__(256) void gram_wmma_kernel(
    const float* __restrict__ H, float* __restrict__ G)
{
    const int wave = (blockIdx.x * 256 + threadIdx.x) >> 5;  // 0..4095
    const int lane = threadIdx.x & 31;
    const int ti = wave >> 6;    // tile row 0..63
    const int tj = wave & 63;    // tile col 0..63
    const int half = lane >> 4;  // 0: K pair {0,1}, 1: K pair {2,3}
    const int lrow = lane & 15;

    const float* rowA = H + (ti * 16 + lrow) * D;
    const float* rowB = H + (tj * 16 + lrow) * D;

    v8f c = {};
#pragma unroll 8
    for (int k0 = 0; k0 < D; k0 += 4) {
        const int ko = k0 + 2 * half;
        v2f a = *(const v2f*)(rowA + ko);
        v2f b = *(const v2f*)(rowB + ko);
        // 8 args: (neg_a, A, neg_b, B, c_mod, C, reuse_a, reuse_b)
        c = __builtin_amdgcn_wmma_f32_16x16x4_f32(
            false, a, false, b, (short)0, c, false, false);
    }

    // C layout: VGPR v -> (M = v + 8*half, N = lrow)
    float* gbase = G + (ti * 16 + half * 8) * N_NODES + tj * 16 + lrow;
#pragma unroll
    for (int v = 0; v < 8; ++v)
        gbase[v * N_NODES] = c[v];
}

// ---------------------------------------------------------------------------
// out[e] = sigmoid( G[src[e]][dst[e]] + bias )
// ---------------------------------------------------------------------------
__global__ __launch_bounds__(256) void edge_kernel(
    const int* __restrict__ ei, const float* __restrict__ G,
    const float* __restrict__ bias, float* __restrict__ out, int E)
{
    const int e = blockIdx.x * 256 + threadIdx.x;
    if (e >= E) return;
    const int s = ei[e];
    const int d = ei[E + e];
    const float logit = G[s * N_NODES + d] + bias[0];
    out[e] = 1.0f / (1.0f + __expf(-logit));
}

extern "C" void kernel_launch(void* const* d_in, const int* in_sizes, int n_in,
                              void* d_out, int out_size, void* d_ws, size_t ws_size,
                              hipStream_t stream)
{
    const float* z    = (const float*)d_in[0];
    const int*   ei   = (const int*)d_in[1];
    const float* W1   = (const float*)d_in[2];
    const float* b1   = (const float*)d_in[3];
    const float* W2   = (const float*)d_in[4];
    const float* b2   = (const float*)d_in[5];
    const float* bias = (const float*)d_in[6];
    float* out = (float*)d_out;
    const int E = in_sizes[1] / 2;

    float* ws = (float*)d_ws;
    float* m1 = ws;                      // 128
    float* m2 = ws + 128;                // 128
    float* h1 = ws + 256;                // 1024*128
    float* h2 = h1 + N_NODES * D;        // 1024*128
    float* G  = h2 + N_NODES * D;        // 1024*1024

    // Layer 1: m1 = mean(z) @ W1 + b1 ; h1 = relu(z + m1)
    mean_matvec_kernel<<<1, 1024, 0, stream>>>(z, W1, b1, m1);
    add_bias_kernel<<<(N_NODES * D) / 256, 256, 0, stream>>>(z, m1, h1, 1);

    // Layer 2: m2 = mean(h1) @ W2 + b2 ; h2 = h1 + m2
    mean_matvec_kernel<<<1, 1024, 0, stream>>>(h1, W2, b2, m2);
    add_bias_kernel<<<(N_NODES * D) / 256, 256, 0, stream>>>(h1, m2, h2, 0);

    // Gram matrix via f32 WMMA: G = h2 @ h2^T
    gram_wmma_kernel<<<512, 256, 0, stream>>>(h2, G);

    // Edge scoring: gather + sigmoid
    edge_kernel<<<(E + 255) / 256, 256, 0, stream>>>(ei, G, bias, out, E);
}